// PCAw_Pool_86285892976675
// MI455X (gfx1250) — compile-verified
//
#include <hip/hip_runtime.h>

typedef __attribute__((ext_vector_type(2))) float v2f;
typedef __attribute__((ext_vector_type(8))) float v8f;

#define B_  8
#define C_  32
#define FH_ 128
#define TW_ 512
#define H_  31
#define W_  127
#define D_  64
#define HW_ (H_ * W_)            // 3937
#define N_  (C_ * HW_)           // 125984 (divisible by 4)
#define NCH_ (N_ / 4)            // 31496 row-chunks of 4
#define BLOCKS_PER_BATCH 8
#define WAVES_PER_BATCH  (BLOCKS_PER_BATCH * 8)   // 64
#define OUT_TOTAL (B_ * C_ * H_ * W_)

// ---------------------------------------------------------------------------
// Kernel 1: per-batch second-moment matrix S = X^T X (64x64) + column sums,
// where X rows are unfolded 8x8/stride-4 patches built on the fly from x.
// Uses V_WMMA_F32_16X16X4_F32: per 4-row chunk, 4 fragments (16 features each)
// feed a 4x4 grid of 16x16 WMMA accumulators covering the full 64x64 output.
// ---------------------------------------------------------------------------
__global__ __launch_bounds__(256) void gram_kernel(const float* __restrict__ x,
                                                   float* __restrict__ S,
                                                   float* __restrict__ colsum) {
    const int batch = blockIdx.x >> 3;
    const int blk   = blockIdx.x & 7;
    const int wave  = threadIdx.x >> 5;
    const int lane  = threadIdx.x & 31;
    const int half  = lane >> 4;      // 0: rows n0,n0+1   1: rows n0+2,n0+3
    const int l     = lane & 15;      // feature-within-tile / column index
    const int waveId = blk * 8 + wave;

    const float* xb = x + (size_t)batch * C_ * FH_ * TW_;

    v8f acc[4][4];
    v8f zero = {};
#pragma unroll
    for (int i = 0; i < 4; ++i)
#pragma unroll
        for (int j = 0; j < 4; ++j)
            acc[i][j] = zero;

    float csum[4] = {0.f, 0.f, 0.f, 0.f};

    for (int ch = waveId; ch < NCH_; ch += WAVES_PER_BATCH) {
        const int n0 = ch * 4 + half * 2;   // first of this lane-half's 2 rows
        const int n1 = n0 + 1;
        int c0 = n0 / HW_;  int rem0 = n0 - c0 * HW_;
        int h0 = rem0 / W_; int w0 = rem0 - h0 * W_;
        int c1 = n1 / HW_;  int rem1 = n1 - c1 * HW_;
        int h1 = rem1 / W_; int w1 = rem1 - h1 * W_;
        const int rb0 = (c0 * FH_ + h0 * 4) * TW_ + w0 * 4;
        const int rb1 = (c1 * FH_ + h1 * 4) * TW_ + w1 * 4;

        v2f frag[4];
#pragma unroll
        for (int t = 0; t < 4; ++t) {
            const int d  = t * 16 + l;     // feature index 0..63
            const int fi = d >> 3;
            const int ti = d & 7;
            const int o  = fi * TW_ + ti;
            float a = xb[rb0 + o];
            float b = xb[rb1 + o];
            frag[t].x = a;
            frag[t].y = b;
            csum[t] += a + b;
        }
        // S_tile(i,j) += A(i)^T-chunk x B(j)-chunk : same fragment serves as
        // both A (16x4) and B (4x16) operand under the 32-bit lane layouts.
#pragma unroll
        for (int i = 0; i < 4; ++i)
#pragma unroll
            for (int j = 0; j < 4; ++j)
                acc[i][j] = __builtin_amdgcn_wmma_f32_16x16x4_f32(
                    false, frag[i], false, frag[j], (short)0, acc[i][j],
                    false, false);
    }

    // Flush accumulators: 16x16 f32 C/D layout -> row d = i*16 + half*8 + v,
    // col e = j*16 + l held in VGPR v of this lane.
    float* Sb = S + batch * D_ * D_;
#pragma unroll
    for (int i = 0; i < 4; ++i)
#pragma unroll
        for (int j = 0; j < 4; ++j)
#pragma unroll
            for (int v = 0; v < 8; ++v) {
                const int d = i * 16 + half * 8 + v;
                const int e = j * 16 + l;
                unsafeAtomicAdd(&Sb[d * D_ + e], acc[i][j][v]);
            }
    float* cs = colsum + batch * D_;
#pragma unroll
    for (int t = 0; t < 4; ++t)
        unsafeAtomicAdd(&cs[t * 16 + l], csum[t]);
}

// ---------------------------------------------------------------------------
// Kernel 2: per batch (1 block, 64 threads): build normalized Gram matrix
// G = diag(1/sig) (S - N mu mu^T) diag(1/sig), Jacobi-eigendecompose it in
// LDS, sort eigenpairs descending, form v = V . softmax(w), normalize, and
// fold sigma/mean into v' and a per-batch constant for the score pass.
// ---------------------------------------------------------------------------
__global__ __launch_bounds__(64) void eigen_kernel(const float* __restrict__ S,
                                                   const float* __restrict__ colsum,
                                                   const float* __restrict__ weights,
                                                   float* __restrict__ vprime,
                                                   float* __restrict__ constb) {
    __shared__ float G[D_][D_ + 1];
    __shared__ float V[D_][D_ + 1];
    __shared__ float mu[D_], isig[D_], wsm[D_], sgn[D_], vv[D_], vpS[D_];
    __shared__ int   perm[D_];
    __shared__ float shc, shs, shn;

    const int b = blockIdx.x;
    const int t = threadIdx.x;
    const float* Sb = S + b * D_ * D_;
    const float invN = 1.0f / (float)N_;

    float m   = colsum[b * D_ + t] * invN;
    float var = Sb[t * D_ + t] * invN - m * m;
    mu[t]   = m;
    isig[t] = rsqrtf(var + 1e-6f);
    __syncthreads();

    for (int d2 = 0; d2 < D_; ++d2) {
        G[d2][t] = (Sb[d2 * D_ + t] - (float)N_ * mu[d2] * mu[t]) *
                   isig[d2] * isig[t];
        V[d2][t] = (d2 == t) ? 1.f : 0.f;
    }
    __syncthreads();

    // Cyclic Jacobi sweeps (G' = J^T G J), rotations applied column-parallel.
    for (int sweep = 0; sweep < 10; ++sweep) {
        for (int p = 0; p < D_ - 1; ++p) {
            for (int q = p + 1; q < D_; ++q) {
                if (t == 0) {
                    float apq = G[p][q];
                    float c = 1.f, s = 0.f;
                    if (fabsf(apq) > 1e-12f) {
                        float tau = (G[q][q] - G[p][p]) / (2.f * apq);
                        float tt  = ((tau >= 0.f) ? 1.f : -1.f) /
                                    (fabsf(tau) + sqrtf(1.f + tau * tau));
                        c = rsqrtf(1.f + tt * tt);
                        s = tt * c;
                    }
                    shc = c; shs = s;
                }
                __syncthreads();
                float c = shc, s = shs;
                float a0 = G[p][t], b0 = G[q][t];   // left: rows p,q
                G[p][t] = c * a0 - s * b0;
                G[q][t] = s * a0 + c * b0;
                __syncthreads();
                a0 = G[t][p]; b0 = G[t][q];         // right: cols p,q
                G[t][p] = c * a0 - s * b0;
                G[t][q] = s * a0 + c * b0;
                a0 = V[t][p]; b0 = V[t][q];         // V = V J
                V[t][p] = c * a0 - s * b0;
                V[t][q] = s * a0 + c * b0;
                __syncthreads();
            }
        }
    }

    if (t == 0) {
        for (int i = 0; i < D_; ++i) perm[i] = i;
        for (int i = 0; i < D_ - 1; ++i) {          // sort eigvals descending
            int best = i;
            for (int j = i + 1; j < D_; ++j)
                if (G[perm[j]][perm[j]] > G[perm[best]][perm[best]]) best = j;
            int tmp = perm[i]; perm[i] = perm[best]; perm[best] = tmp;
        }
        float mx = weights[0];                      // softmax(weights)
        for (int i = 1; i < D_; ++i) mx = fmaxf(mx, weights[i]);
        float sum = 0.f;
        for (int i = 0; i < D_; ++i) { wsm[i] = __expf(weights[i] - mx); sum += wsm[i]; }
        float inv = 1.f / sum;
        for (int i = 0; i < D_; ++i) wsm[i] *= inv;
    }
    __syncthreads();

    {   // deterministic sign convention: largest-|.| component positive
        int col = perm[t];
        float best = 0.f, val = 1.f;
        for (int d2 = 0; d2 < D_; ++d2) {
            float v0 = V[d2][col];
            if (fabsf(v0) > best) { best = fabsf(v0); val = v0; }
        }
        sgn[t] = (val < 0.f) ? -1.f : 1.f;
    }
    __syncthreads();

    float vd = 0.f;
    for (int e = 0; e < D_; ++e)
        vd += wsm[e] * sgn[e] * V[t][perm[e]];
    vv[t] = vd;
    __syncthreads();
    if (t == 0) {
        float ss = 0.f;
        for (int i = 0; i < D_; ++i) ss += vv[i] * vv[i];
        shn = sqrtf(ss) + 1e-8f;
    }
    __syncthreads();
    float vp = (vv[t] / shn) * isig[t];   // fold 1/sigma into v
    vpS[t] = vp;
    vprime[b * D_ + t] = vp;
    __syncthreads();
    if (t == 0) {
        float cb = 0.f;                   // fold mean subtraction into a const
        for (int i = 0; i < D_; ++i) cb += mu[i] * vpS[i];
        constb[b] = cb;
    }
}

// ---------------------------------------------------------------------------
// Kernel 3: scores(b,c,h,w) = sum_{fi,ti} x[b,c,4h+fi,4w+ti] * v'[fi*8+ti]
//                             - const_b        (vectorized float4 loads)
// ---------------------------------------------------------------------------
__global__ __launch_bounds__(256) void scores_kernel(const float* __restrict__ x,
                                                     const float* __restrict__ vprime,
                                                     const float* __restrict__ constb,
                                                     float* __restrict__ out) {
    int idx = blockIdx.x * 256 + threadIdx.x;
    if (idx >= OUT_TOTAL) return;
    int w  = idx % W_;  int t1 = idx / W_;
    int h  = t1 % H_;   int t2 = t1 / H_;
    int c  = t2 % C_;   int b  = t2 / C_;
    const float* xp = x + ((size_t)(b * C_ + c) * FH_ + h * 4) * TW_ + w * 4;
    const float* vp = vprime + b * D_;
    float s = 0.f;
#pragma unroll
    for (int fi = 0; fi < 8; ++fi) {
        const float4 a  = *(const float4*)(xp + fi * TW_);
        const float4 bb = *(const float4*)(xp + fi * TW_ + 4);
        const float4 va = *(const float4*)(vp + fi * 8);
        const float4 vb = *(const float4*)(vp + fi * 8 + 4);
        s += a.x * va.x + a.y * va.y + a.z * va.z + a.w * va.w +
             bb.x * vb.x + bb.y * vb.y + bb.z * vb.z + bb.w * vb.w;
    }
    out[idx] = s - constb[b];
}

extern "C" void kernel_launch(void* const* d_in, const int* in_sizes, int n_in,
                              void* d_out, int out_size, void* d_ws, size_t ws_size,
                              hipStream_t stream) {
    const float* x       = (const float*)d_in[0];
    const float* weights = (const float*)d_in[1];
    float* out = (float*)d_out;

    float* S      = (float*)d_ws;              // 8*64*64
    float* colsum = S + B_ * D_ * D_;          // 8*64
    float* vprime = colsum + B_ * D_;          // 8*64
    float* constb = vprime + B_ * D_;          // 8

    hipMemsetAsync(d_ws, 0,
                   (size_t)(B_ * D_ * D_ + B_ * D_) * sizeof(float), stream);

    gram_kernel<<<dim3(B_ * BLOCKS_PER_BATCH), dim3(256), 0, stream>>>(x, S, colsum);
    eigen_kernel<<<dim3(B_), dim3(64), 0, stream>>>(S, colsum, weights, vprime, constb);
    scores_kernel<<<dim3((OUT_TOTAL + 255) / 256), dim3(256), 0, stream>>>(x, vprime, constb, out);
}